// HGSA_layer_12403865551354
// MI455X (gfx1250) — compile-verified
//
#include <hip/hip_runtime.h>

typedef __attribute__((ext_vector_type(2))) float v2f;
typedef __attribute__((ext_vector_type(8))) float v8f;

#define HEADS 4
#define OUTD  32
#define FDIM  128   // HEADS*OUTD
#define NEG_SLOPE 0.2f

#if __has_builtin(__builtin_amdgcn_s_wait_asynccnt)
#define HAVE_ASYNC 1
#else
#define HAVE_ASYNC 0
#endif

#define LDS_AS __attribute__((address_space(3)))

// ---------------- zero init (emax/den/agg region) ----------------
__global__ void hgsa_zero(unsigned* __restrict__ p, long long nwords) {
    long long t = (long long)blockIdx.x * blockDim.x + threadIdx.x;
    if (t < nwords) p[t] = 0u;
}

// ---------------- el[n,h] = dot(feat[n, h*32: ], attn_l[h]) ----------------
__global__ void hgsa_el(const float* __restrict__ feat, const float* __restrict__ attn_l,
                        float* __restrict__ el, int N4) {
    int t = blockIdx.x * blockDim.x + threadIdx.x;
    if (t >= N4) return;
    int n = t >> 2, h = t & 3;
    const float* f = feat + (size_t)n * FDIM + h * OUTD;
    const float* a = attn_l + h * OUTD;
    float s = 0.f;
#pragma unroll
    for (int d = 0; d < OUTD; ++d) s += f[d] * a[d];
    el[t] = s;
}

// ---------------- e[e,h] = leakyrelu(el[src]+em), atomic segment max ----------------
__global__ void hgsa_edge(const float* __restrict__ edge_feat, const float* __restrict__ attn_m,
                          const float* __restrict__ el, const int* __restrict__ src,
                          const int* __restrict__ dst, float* __restrict__ ebuf,
                          unsigned* __restrict__ emax, int E) {
    int e = blockIdx.x * blockDim.x + threadIdx.x;
    if (e >= E) return;
    const float* ef = edge_feat + (size_t)e * 32;
    float em0 = 0.f, em1 = 0.f, em2 = 0.f, em3 = 0.f;
#pragma unroll
    for (int d = 0; d < 32; ++d) {
        float v = ef[d];
        em0 += v * attn_m[d];
        em1 += v * attn_m[32 + d];
        em2 += v * attn_m[64 + d];
        em3 += v * attn_m[96 + d];
    }
    float em[4] = {em0, em1, em2, em3};
    int s = src[e], dd = dst[e];
#pragma unroll
    for (int h = 0; h < HEADS; ++h) {
        float x = el[s * 4 + h] + em[h];
        x = x > 0.f ? x : NEG_SLOPE * x;
        ebuf[(size_t)e * 4 + h] = x;
        unsigned u = __float_as_uint(x);
        u = (u & 0x80000000u) ? ~u : (u | 0x80000000u);
        atomicMax(&emax[dd * 4 + h], u);
    }
}

// ---------------- ex = exp(e - max), atomic segment sum of denominators ----------------
__global__ void hgsa_ex(float* __restrict__ ebuf, const unsigned* __restrict__ emax,
                        const int* __restrict__ dst, float* __restrict__ den, int E4) {
    int t = blockIdx.x * blockDim.x + threadIdx.x;
    if (t >= E4) return;
    int e = t >> 2, h = t & 3;
    int d = dst[e];
    unsigned u = emax[d * 4 + h];
    float mx = (u & 0x80000000u) ? __uint_as_float(u & 0x7fffffffu) : __uint_as_float(~u);
    float ex = __expf(ebuf[t] - mx);
    ebuf[t] = ex;
    atomicAdd(&den[d * 4 + h], ex);
}

// ---------------- agg[dst, h, d] += feat[src, h, d] * a[e,h]  (wave per edge) ----------------
__global__ void hgsa_scatter(const float* __restrict__ feat, const float* __restrict__ exbuf,
                             const float* __restrict__ den, const int* __restrict__ src,
                             const int* __restrict__ dst, float* __restrict__ agg, int E) {
    int warp = (int)(((long long)blockIdx.x * blockDim.x + threadIdx.x) >> 5);
    int lane = threadIdx.x & 31;
    if (warp >= E) return;
    int s = src[warp], d = dst[warp];
#pragma unroll
    for (int h = 0; h < HEADS; ++h) {
        float a = exbuf[(size_t)warp * 4 + h] / den[d * 4 + h];
        float v = feat[(size_t)s * FDIM + h * OUTD + lane] * a;
        atomicAdd(&agg[(size_t)d * FDIM + h * OUTD + lane], v);
    }
}

// ======================= GEMM: out[N,128] = H[N,K] @ agg[K,128] =======================
// Block: 256 threads = 8 wave32 arranged 2x4. Block tile: 32 rows x 128 cols.
// Wave (wrow, wcol) computes 16 rows x 32 cols -> two 16x16 WMMA accumulators sharing A.
// K consumed in chunks of KC=32, double-buffered via async global->LDS DMA (ASYNCcnt).
constexpr int KC   = 32;
constexpr int KCP  = KC + 4;    // sH stride 36: banks r*36%64 distinct for r<16, halves disjoint
constexpr int BST  = FDIM + 8;  // sB stride 136: K+2 rows land +16 banks away -> halves disjoint

#if HAVE_ASYNC
// 16-byte async copy global -> LDS, tracked by ASYNCcnt (inline asm: portable across
// toolchains, bypasses the builtin's unspellable cuda_device address space).
__device__ __forceinline__ void async_cp16(const float* g, float* l) {
    unsigned lds_off = (unsigned)(unsigned long long)(LDS_AS float*)l;
    asm volatile("global_load_async_to_lds_b128 %0, %1, off"
                 :: "v"(lds_off), "v"(g)
                 : "memory");
}

// Every thread issues exactly 1 (sH) + 4 (sB) b128 copies -> 5 async instrs per wave.
__device__ __forceinline__ void issue_chunk(const float* __restrict__ Hm,
                                            const float* __restrict__ agg,
                                            float (*sH)[KCP], float (*sB)[BST],
                                            int row0, int k0, int K, int Nrows, int tid) {
    // sH: 32 rows x 32 cols = 256 x 16B chunks (threads 0-7 cover one row: coalesced)
    {
        int r  = tid >> 3;              // 0..31
        int c4 = (tid & 7) << 2;        // 0,4,...,28
        int rr = row0 + r; rr = rr < Nrows ? rr : Nrows - 1;   // clamp (N%32==0 in practice)
        async_cp16(&Hm[(size_t)rr * K + k0 + c4], &sH[r][c4]);
    }
    // sB: 32 rows x 128 cols = 1024 x 16B chunks, 4 per thread, coalesced per row
#pragma unroll
    for (int i = 0; i < 4; ++i) {
        int idx = tid + (i << 8);
        int r   = idx >> 5;             // 0..31
        int c4  = (idx & 31) << 2;      // 0..124
        async_cp16(&agg[(size_t)(k0 + r) * FDIM + c4], &sB[r][c4]);
    }
}
#endif

// synchronous zero-padded fill (fallback path + K tail)
__device__ __forceinline__ void sync_fill(const float* __restrict__ Hm,
                                          const float* __restrict__ agg,
                                          float (*sH)[KCP], float (*sB)[BST],
                                          int row0, int k0, int kc, int K, int Nrows, int tid) {
    for (int idx = tid; idx < 32 * KC; idx += 256) {
        int r = idx >> 5, c = idx & (KC - 1);
        float v = 0.f;
        if (c < kc && (row0 + r) < Nrows) v = Hm[(size_t)(row0 + r) * K + k0 + c];
        sH[r][c] = v;
    }
    for (int idx = tid; idx < KC * FDIM; idx += 256) {
        int r = idx >> 7, c = idx & 127;
        sB[r][c] = (r < kc) ? agg[(size_t)(k0 + r) * FDIM + c] : 0.f;
    }
}

__device__ __forceinline__ void gemm_chunk(const float (*sH)[KCP], const float (*sB)[BST],
                                           v8f& acc0, v8f& acc1,
                                           int arow, int cb0, int half, int l16) {
#pragma unroll
    for (int ks = 0; ks < KC; ks += 4) {
        int kk = ks + 2 * half;         // lane half selects K pair (ISA 7.12.2 fp32 layout)
        v2f a, b0, b1;
        a.x  = sH[arow][kk];            a.y  = sH[arow][kk + 1];
        b0.x = sB[kk][cb0 + l16];       b0.y = sB[kk + 1][cb0 + l16];
        b1.x = sB[kk][cb0 + 16 + l16];  b1.y = sB[kk + 1][cb0 + 16 + l16];
        acc0 = __builtin_amdgcn_wmma_f32_16x16x4_f32(false, a, false, b0,
                                                     (short)0, acc0, false, false);
        acc1 = __builtin_amdgcn_wmma_f32_16x16x4_f32(false, a, false, b1,
                                                     (short)0, acc1, false, false);
    }
}

__global__ __launch_bounds__(256) void hgsa_gemm(const float* __restrict__ Hm,
                                                 const float* __restrict__ agg,
                                                 float* __restrict__ out,
                                                 int Nrows, int K) {
    __shared__ float sH[2][32][KCP];
    __shared__ float sB[2][KC][BST];

    const int tid  = threadIdx.x;
    const int row0 = blockIdx.x * 32;
    const int wave = tid >> 5;
    const int lane = tid & 31;
    const int wrow = wave >> 2;          // 0..1
    const int wcol = wave & 3;           // 0..3
    const int half = lane >> 4;
    const int l16  = lane & 15;
    const int arow = 16 * wrow + l16;
    const int cb0  = 32 * wcol;

    v8f acc0 = {}, acc1 = {};

    const int nfull = K / KC;
    const int tail  = K - nfull * KC;    // multiple of 4

#if HAVE_ASYNC
    if (nfull > 0)
        issue_chunk(Hm, agg, sH[0], sB[0], row0, 0, K, Nrows, tid);
    int buf = 0;
    for (int ci = 0; ci < nfull; ++ci) {
        if (ci + 1 < nfull) {
            issue_chunk(Hm, agg, sH[buf ^ 1], sB[buf ^ 1], row0, (ci + 1) * KC, K, Nrows, tid);
            __builtin_amdgcn_s_wait_asynccnt(5);   // 5 outstanding = next chunk only
        } else {
            __builtin_amdgcn_s_wait_asynccnt(0);
        }
        __syncthreads();
        gemm_chunk(sH[buf], sB[buf], acc0, acc1, arow, cb0, half, l16);
        __syncthreads();
        buf ^= 1;
    }
#else
    int buf = 0;
    for (int ci = 0; ci < nfull; ++ci) {
        sync_fill(Hm, agg, sH[buf], sB[buf], row0, ci * KC, KC, K, Nrows, tid);
        __syncthreads();
        gemm_chunk(sH[buf], sB[buf], acc0, acc1, arow, cb0, half, l16);
        __syncthreads();
        buf ^= 1;
    }
#endif
    if (tail > 0) {
        sync_fill(Hm, agg, sH[buf], sB[buf], row0, nfull * KC, tail, K, Nrows, tid);
        __syncthreads();
        gemm_chunk(sH[buf], sB[buf], acc0, acc1, arow, cb0, half, l16);
        __syncthreads();
    }

    // C/D 16x16 fp32 layout: VGPR i -> row (i + 8*half), col = l16
#pragma unroll
    for (int i = 0; i < 8; ++i) {
        int r = row0 + 16 * wrow + i + 8 * half;
        if (r < Nrows) {
            out[(size_t)r * FDIM + cb0 + l16]      = acc0[i];
            out[(size_t)r * FDIM + cb0 + 16 + l16] = acc1[i];
        }
    }
}

extern "C" void kernel_launch(void* const* d_in, const int* in_sizes, int n_in,
                              void* d_out, int out_size, void* d_ws, size_t ws_size,
                              hipStream_t stream) {
    const float* feat      = (const float*)d_in[0];
    const float* edge_feat = (const float*)d_in[1];
    const float* Hm        = (const float*)d_in[2];
    const float* attn_l    = (const float*)d_in[3];
    const float* attn_m    = (const float*)d_in[4];
    const int*   src       = (const int*)d_in[5];
    const int*   dst       = (const int*)d_in[6];
    float* out = (float*)d_out;

    const int N = in_sizes[0] / FDIM;                 // 20000
    const int E = in_sizes[5];                        // 320000
    const int M = (int)((long long)in_sizes[2] / N);  // 5000

    char* ws = (char*)d_ws;
    size_t off = 0;
    auto carve = [&](size_t bytes) -> void* {
        void* p = ws + off;
        off = (off + bytes + 255) & ~(size_t)255;
        return p;
    };
    float*    ebuf = (float*)   carve((size_t)E * 4 * sizeof(float));
    size_t zbeg = off;
    unsigned* emax = (unsigned*)carve((size_t)M * 4 * sizeof(unsigned));
    float*    den  = (float*)   carve((size_t)M * 4 * sizeof(float));
    float*    agg  = (float*)   carve((size_t)M * FDIM * sizeof(float));
    size_t zend = off;
    float*    el   = (float*)   carve((size_t)N * 4 * sizeof(float));
    (void)ws_size; (void)n_in; (void)out_size;

    long long zwords = (long long)(zend - zbeg) / 4;
    hgsa_zero<<<(int)((zwords + 255) / 256), 256, 0, stream>>>((unsigned*)(ws + zbeg), zwords);

    int N4 = N * 4;
    hgsa_el<<<(N4 + 255) / 256, 256, 0, stream>>>(feat, attn_l, el, N4);

    hgsa_edge<<<(E + 255) / 256, 256, 0, stream>>>(edge_feat, attn_m, el, src, dst, ebuf, emax, E);

    int E4 = E * 4;
    hgsa_ex<<<(E4 + 255) / 256, 256, 0, stream>>>(ebuf, emax, dst, den, E4);

    long long sthreads = (long long)E * 32;
    hgsa_scatter<<<(int)((sthreads + 255) / 256), 256, 0, stream>>>(feat, ebuf, den, src, dst, agg, E);

    hgsa_gemm<<<(N + 31) / 32, 256, 0, stream>>>(Hm, agg, out, N, M);
}